// SliceFastCUDALatticeModule_25400436588645
// MI455X (gfx1250) — compile-verified
//
#include <hip/hip_runtime.h>
#include <hip/hip_bf16.h>

#define NV_   300000
#define P_    500000
#define VD_   64
#define NC_   20
#define EPSF  1e-5f

typedef float v2f __attribute__((ext_vector_type(2)));
typedef float v8f __attribute__((ext_vector_type(8)));

__device__ __forceinline__ v8f v8f_zero() {
    v8f z = {0.f,0.f,0.f,0.f,0.f,0.f,0.f,0.f};
    return z;
}

// D = A(16x4 f32) * B(4x16 f32) + C(16x16 f32), wave32 striped
__device__ __forceinline__ v8f wmma4(v2f a, v2f b, v8f c) {
    return __builtin_amdgcn_wmma_f32_16x16x4_f32(false, a, false, b, (short)0, c, false, false);
}

// ---------------------------------------------------------------- utilities
__global__ void zero_kernel(float* __restrict__ p, int n) {
    int i = blockIdx.x * blockDim.x + threadIdx.x;
    if (i < n) p[i] = 0.0f;
}

// per-channel sum / sumsq of lv [NV, 64] -> stats[0:64]=sum, stats[64:128]=sumsq
__global__ __launch_bounds__(256)
void stats0_kernel(const float* __restrict__ lv, float* __restrict__ stats) {
    __shared__ float ls[64], lq[64];
    int t = threadIdx.x;
    if (t < 64) { ls[t] = 0.0f; lq[t] = 0.0f; }
    __syncthreads();
    size_t gtid   = (size_t)blockIdx.x * blockDim.x + t;
    size_t stride = (size_t)gridDim.x * blockDim.x;   // multiple of 64
    int c = (int)(gtid & 63);
    float s = 0.0f, q = 0.0f;
    for (size_t i = gtid; i < (size_t)NV_ * 64; i += stride) {
        float v = lv[i];
        s += v; q += v * v;
    }
    atomicAdd(&ls[c], s);
    atomicAdd(&lq[c], q);
    __syncthreads();
    if (t < 64) { atomicAdd(&stats[t], ls[t]); atomicAdd(&stats[64 + t], lq[t]); }
}

// fold GN stats into per-channel scale/bias:  xn = x*scale + bias
__global__ void finalize_gn_kernel(const float* __restrict__ stats,
                                   const float* __restrict__ gnw,
                                   const float* __restrict__ gnb,
                                   float* __restrict__ sb,
                                   int nch, int chpergroup, float inv_count) {
    int c = threadIdx.x;
    if (c < nch) {
        int g = c / chpergroup;
        float s = 0.0f, q = 0.0f;
        for (int j = 0; j < chpergroup; ++j) {
            s += stats[g * chpergroup + j];
            q += stats[nch + g * chpergroup + j];
        }
        float mean = s * inv_count;
        float var  = q * inv_count - mean * mean;
        float sc   = gnw[c] * rsqrtf(var + EPSF);
        sb[c]        = sc;
        sb[nch + c]  = gnb[c] - mean * sc;
    }
}

// ------------------------------------------- fused GN->ReLU->GEMM (+ next-GN stats)
// out[m][n] = sum_k relu(in[m][k]*scale[k]+bias[k]) * W[n][k]
// CIN: input channels (mult of 4); WROWS: real output channels; NT: N-tiles (NT*16 >= WROWS)
// STATS_CH: per-channel stats of the output to accumulate (0 = none)
template<int CIN, int WROWS, int NT, int STATS_CH>
__global__ __launch_bounds__(256)
void gn_gemm_kernel(const float* __restrict__ in, const float* __restrict__ W,
                    const float* __restrict__ sb, float* __restrict__ out,
                    float* __restrict__ stats, int nrows) {
    __shared__ float lw[NT * 16][CIN];
    __shared__ float lsc[CIN], lbi[CIN];
    __shared__ float lss[(STATS_CH > 0) ? 2 * STATS_CH : 1];

    int t = threadIdx.x;
    for (int i = t; i < NT * 16 * CIN; i += 256) {
        int r = i / CIN, k = i % CIN;
        lw[r][k] = (r < WROWS) ? W[r * CIN + k] : 0.0f;   // zero-pad N
    }
    for (int i = t; i < CIN; i += 256) { lsc[i] = sb[i]; lbi[i] = sb[CIN + i]; }
    if constexpr (STATS_CH > 0) {
        for (int i = t; i < 2 * STATS_CH; i += 256) lss[i] = 0.0f;
    }
    __syncthreads();

    int wave = t >> 5, lane = t & 31;
    int h = lane >> 4, mi = lane & 15;
    int mbase = blockIdx.x * 128 + wave * 16;
    int m = mbase + mi;
    float okf = (m < nrows) ? 1.0f : 0.0f;
    const float* arow = in + (size_t)((m < nrows) ? m : 0) * CIN;

    v8f acc[NT];
#pragma unroll
    for (int nt = 0; nt < NT; ++nt) acc[nt] = v8f_zero();

#pragma unroll
    for (int kk = 0; kk < CIN / 4; ++kk) {
        int k0 = kk * 4 + 2 * h;                 // A frag: lane holds K = 2h+{0,1}
        v2f g = *(const v2f*)(arow + k0);        // global_load_b64, row resident in L2
        v2f a;
        a[0] = fmaxf(g[0] * lsc[k0]     + lbi[k0],     0.0f) * okf;
        a[1] = fmaxf(g[1] * lsc[k0 + 1] + lbi[k0 + 1], 0.0f) * okf;
#pragma unroll
        for (int nt = 0; nt < NT; ++nt) {
            v2f b;                                // B[k][n] = W[n][k]
            b[0] = lw[nt * 16 + mi][k0];
            b[1] = lw[nt * 16 + mi][k0 + 1];
            acc[nt] = wmma4(a, b, acc[nt]);
        }
    }

#pragma unroll
    for (int nt = 0; nt < NT; ++nt) {
        int n = nt * 16 + mi;
        float s = 0.0f, q = 0.0f;
#pragma unroll
        for (int r = 0; r < 8; ++r) {
            int mm = mbase + r + 8 * h;          // C layout: M = r + 8*half
            float val = acc[nt][r];
            if (mm < nrows && n < WROWS) {
                out[(size_t)mm * WROWS + n] = val;
                s += val; q += val * val;
            }
        }
        if constexpr (STATS_CH > 0) {
            if (n < STATS_CH) {
                atomicAdd(&lss[n], s);
                atomicAdd(&lss[STATS_CH + n], q);
            }
        }
    }
    if constexpr (STATS_CH > 0) {
        __syncthreads();
        if (t < 2 * STATS_CH) atomicAdd(&stats[t], lss[t]);
    }
}

// ------------------------------- gather + max-normalize + delta-w + slice + classify
__global__ __launch_bounds__(256)
void slice_classify_kernel(const float* __restrict__ lv, const int* __restrict__ idx,
                           const float* __restrict__ bary, const float* __restrict__ x3,
                           const float* __restrict__ gamma, const float* __restrict__ beta,
                           const float* __restrict__ dW_W, const float* __restrict__ dW_b,
                           const float* __restrict__ cls_W, const float* __restrict__ cls_b,
                           float* __restrict__ out) {
    __shared__ float lcw[32][VD_];                 // classify W, N padded to 32
    __shared__ float lcb[32];
    __shared__ float lgm[9], lbt[9], ldw[9];
    __shared__ float ldwb_s;
    __shared__ __align__(16) int   lidx[128][4];
    __shared__ __align__(16) float lwt[128][4];

    int t = threadIdx.x;
    for (int i = t; i < 32 * VD_; i += 256) {
        int r = i >> 6, k = i & 63;
        lcw[r][k] = (r < NC_) ? cls_W[r * VD_ + k] : 0.0f;
    }
    if (t < 32) lcb[t] = (t < NC_) ? cls_b[t] : 0.0f;
    if (t < 9)  { lgm[t] = gamma[t]; lbt[t] = beta[t]; ldw[t] = dW_W[t]; }
    if (t == 0) ldwb_s = dW_b[0];
    __syncthreads();

    int pbase = blockIdx.x * 128;

    // phase 1: one thread per position -> simplex weights into LDS
    if (t < 128) {
        int p = pbase + t;
        if (p < P_) {
            int4   vi = *(const int4*)(idx + (size_t)p * 4);
            float4 vb = *(const float4*)(bary + (size_t)p * 4);
            int   vis[4] = {vi.x, vi.y, vi.z, vi.w};
            float vbs[4] = {vb.x, vb.y, vb.z, vb.w};
            float g[4][8], mx[9];
#pragma unroll
            for (int j = 0; j < 9; ++j) mx[j] = -3.4e38f;
#pragma unroll
            for (int v = 0; v < 4; ++v) {
                const float* xr = x3 + (size_t)vis[v] * 8;
#pragma unroll
                for (int j = 0; j < 8; ++j) { g[v][j] = xr[j]; mx[j] = fmaxf(mx[j], g[v][j]); }
                mx[8] = fmaxf(mx[8], vbs[v]);
            }
            float sub[9];
#pragma unroll
            for (int j = 0; j < 9; ++j) sub[j] = lgm[j] * mx[j] + lbt[j];
#pragma unroll
            for (int v = 0; v < 4; ++v) {
                float dw = ldwb_s;
#pragma unroll
                for (int j = 0; j < 8; ++j) dw += (g[v][j] - sub[j]) * ldw[j];
                dw += (vbs[v] - sub[8]) * ldw[8];
                lwt[t][v]  = vbs[v] + dw;
                lidx[t][v] = vis[v];
            }
        } else {
#pragma unroll
            for (int v = 0; v < 4; ++v) { lwt[t][v] = 0.0f; lidx[t][v] = 0; }
        }
    }
    __syncthreads();

    // phase 2: per wave, 16 positions; sliced[p][k] built on the fly, classify via WMMA
    int wave = t >> 5, lane = t & 31;
    int h = lane >> 4, mi = lane & 15;
    int lp = wave * 16 + mi;
    int4   vi = *(const int4*)&lidx[lp][0];
    float4 vw = *(const float4*)&lwt[lp][0];
    const float* r0 = lv + (size_t)vi.x * VD_;
    const float* r1 = lv + (size_t)vi.y * VD_;
    const float* r2 = lv + (size_t)vi.z * VD_;
    const float* r3 = lv + (size_t)vi.w * VD_;

    v8f c0 = v8f_zero(), c1 = v8f_zero();
#pragma unroll
    for (int kk = 0; kk < 16; ++kk) {
        int k0 = kk * 4 + 2 * h;
        v2f g0 = *(const v2f*)(r0 + k0);          // lv gathers: L2-resident (76.8MB < 192MB)
        v2f g1 = *(const v2f*)(r1 + k0);
        v2f g2 = *(const v2f*)(r2 + k0);
        v2f g3 = *(const v2f*)(r3 + k0);
        v2f a;
        a[0] = vw.x * g0[0] + vw.y * g1[0] + vw.z * g2[0] + vw.w * g3[0];
        a[1] = vw.x * g0[1] + vw.y * g1[1] + vw.z * g2[1] + vw.w * g3[1];
        v2f b0, b1;                               // B[k][n] = cls_W[n][k]
        b0[0] = lcw[mi][k0];      b0[1] = lcw[mi][k0 + 1];
        b1[0] = lcw[16 + mi][k0]; b1[1] = lcw[16 + mi][k0 + 1];
        c0 = wmma4(a, b0, c0);
        c1 = wmma4(a, b1, c1);
    }

#pragma unroll
    for (int r = 0; r < 8; ++r) {
        int p = pbase + wave * 16 + r + 8 * h;
        if (p < P_) {
            out[(size_t)p * NC_ + mi] = c0[r] + lcb[mi];
            if (mi < 4) out[(size_t)p * NC_ + 16 + mi] = c1[r] + lcb[16 + mi];
        }
    }
}

// ---------------------------------------------------------------------------
extern "C" void kernel_launch(void* const* d_in, const int* in_sizes, int n_in,
                              void* d_out, int out_size, void* d_ws, size_t ws_size,
                              hipStream_t stream) {
    const float* lv     = (const float*)d_in[0];
    // d_in[1] = positions (unused by the reference)
    const int*   idx    = (const int*)  d_in[2];
    const float* bary   = (const float*)d_in[3];
    const float* gn0_w  = (const float*)d_in[4];
    const float* gn0_b  = (const float*)d_in[5];
    const float* lin0_W = (const float*)d_in[6];
    const float* gn1_w  = (const float*)d_in[7];
    const float* gn1_b  = (const float*)d_in[8];
    const float* lin1_W = (const float*)d_in[9];
    const float* gn2_w  = (const float*)d_in[10];
    const float* gn2_b  = (const float*)d_in[11];
    const float* lin2_W = (const float*)d_in[12];
    const float* gamma  = (const float*)d_in[13];
    const float* beta   = (const float*)d_in[14];
    const float* dW_W   = (const float*)d_in[15];
    const float* dW_b   = (const float*)d_in[16];
    const float* cls_W  = (const float*)d_in[17];
    const float* cls_b  = (const float*)d_in[18];
    float* out = (float*)d_out;

    float* ws  = (float*)d_ws;
    float* st0 = ws;            // 128 floats: sum[64], sq[64]
    float* st1 = ws + 128;      // 128
    float* st2 = ws + 256;      // 64
    float* sb0 = ws + 384;      // scale[64], bias[64]
    float* sb1 = ws + 512;      // 128
    float* sb2 = ws + 640;      // 64
    float* x1  = ws + 1024;                    // [NV,64]
    float* x2  = x1 + (size_t)NV_ * 64;        // [NV,32]
    float* x3  = x2 + (size_t)NV_ * 32;        // [NV,8]

    const int rowBlocks = (NV_ + 127) / 128;   // 2344
    const int posBlocks = (P_ + 127) / 128;    // 3907

    zero_kernel<<<4, 256, 0, stream>>>(ws, 1024);
    stats0_kernel<<<512, 256, 0, stream>>>(lv, st0);
    finalize_gn_kernel<<<1, 64, 0, stream>>>(st0, gn0_w, gn0_b, sb0, 64, 2, 1.0f / (2.0f * NV_));
    gn_gemm_kernel<64, 64, 4, 64><<<rowBlocks, 256, 0, stream>>>(lv, lin0_W, sb0, x1, st1, NV_);
    finalize_gn_kernel<<<1, 64, 0, stream>>>(st1, gn1_w, gn1_b, sb1, 64, 2, 1.0f / (2.0f * NV_));
    gn_gemm_kernel<64, 32, 2, 32><<<rowBlocks, 256, 0, stream>>>(x1, lin1_W, sb1, x2, st2, NV_);
    finalize_gn_kernel<<<1, 64, 0, stream>>>(st2, gn2_w, gn2_b, sb2, 32, 1, 1.0f / (1.0f * NV_));
    gn_gemm_kernel<32, 8, 1, 0><<<rowBlocks, 256, 0, stream>>>(x2, lin2_W, sb2, x3, nullptr, NV_);
    slice_classify_kernel<<<posBlocks, 256, 0, stream>>>(lv, idx, bary, x3, gamma, beta,
                                                         dW_W, dW_b, cls_W, cls_b, out);
}